// WaveCell_79319456022537
// MI455X (gfx1250) — compile-verified
//
#include <hip/hip_runtime.h>
#include <cstdint>
#include <cstddef>

// Elastic wave (seistorch) time step for MI455X / gfx1250.
//
// Memory-bound stencil (AI ~1 flop/byte << machine balance at 23.3 TB/s), so
// the optimization targets are data movement, not matrix ops:
//  * two-pass structure; 48MB stress intermediate stays resident in 192MB L2
//  * one (b,z) row of 1024 floats per 256-thread block, float4 per thread
//  * CDNA5 async-to-LDS (ASYNCcnt) staging for rows read with x-neighbors
//  * x wrap via &(NX-1) into LDS, z wrap via row index math (no divergence)

#define NZC 1024
#define NXC 1024

// ---- CDNA5 async global->LDS helpers (gfx1250 VGLOBAL async ops) -----------
// dst LDS byte address in a VGPR (VDST), 64-bit global VA in a VGPR pair.
// The third operand (%2, unused in the template) passes the LDS base pointer
// into the asm so the shared array ESCAPES: with the "memory" clobber the
// compiler must then assume the asm writes LDS, keeping the ds_load consumers
// alive (without it, alias analysis dead-codes the whole kernel body).
__device__ __forceinline__ void async_load_b128(unsigned lds_byte, const float* g,
                                                float* lds_base) {
    unsigned long long ga = (unsigned long long)(uintptr_t)g;
    asm volatile("global_load_async_to_lds_b128 %0, %1, off"
                 :: "v"(lds_byte), "v"(ga), "v"(lds_base)
                 : "memory");
}
__device__ __forceinline__ void async_wait0() {
    asm volatile("s_wait_asynccnt 0x0" ::: "memory");
}

// ---------------------------------------------------------------------------
// Pass 1: stress update.
//   vx_x = vx[z,x]-vx[z,x-1]   vz_z = vz[z+1,x]-vz[z,x]
//   vx_z = vx[z+1,x]-vx[z,x]   vz_x = vz[z,x+1]-vz[z,x]
// Writes y_txx -> out+2S, y_tzz -> out+3S, y_txz -> out+4S.
__global__ __launch_bounds__(256)
void stress_kernel(const float* __restrict__ vp,  const float* __restrict__ vs,
                   const float* __restrict__ rho,
                   const float* __restrict__ vx,  const float* __restrict__ vz,
                   const float* __restrict__ txx, const float* __restrict__ tzz,
                   const float* __restrict__ txz,
                   const float* __restrict__ dtp, const float* __restrict__ hp,
                   const float* __restrict__ d,
                   float* __restrict__ out, int S) {
    __shared__ float sm[4 * NXC];   // [vx row z | vx row z+1 | vz row z | vz row z+1]

    const int tid = threadIdx.x;            // 0..255
    const int z   = blockIdx.x & (NZC - 1);
    const int b   = blockIdx.x >> 10;
    const int x0  = tid << 2;               // 4 floats per thread
    const int zp  = (z + 1) & (NZC - 1);

    const size_t base_z  = ((size_t)b * NZC + z)  * NXC;
    const size_t base_zp = ((size_t)b * NZC + zp) * NXC;

    // Async-stage the four velocity rows this block needs (x-neighbor reuse).
    async_load_b128((unsigned)((0 * NXC + x0) * 4), vx + base_z  + x0, sm);
    async_load_b128((unsigned)((1 * NXC + x0) * 4), vx + base_zp + x0, sm);
    async_load_b128((unsigned)((2 * NXC + x0) * 4), vz + base_z  + x0, sm);
    async_load_b128((unsigned)((3 * NXC + x0) * 4), vz + base_zp + x0, sm);

    // Center-only operands: plain coalesced B128 loads (overlap with async).
    const size_t m = (size_t)z * NXC + x0;
    const float4 vp4  = *(const float4*)(vp  + m);
    const float4 vs4  = *(const float4*)(vs  + m);
    const float4 rh4  = *(const float4*)(rho + m);
    const float4 d4   = *(const float4*)(d   + m);
    const float4 txx4 = *(const float4*)(txx + base_z + x0);
    const float4 tzz4 = *(const float4*)(tzz + base_z + x0);
    const float4 txz4 = *(const float4*)(txz + base_z + x0);
    const float  dt   = *dtp;
    const float  dth  = dt / *hp;

    async_wait0();
    __syncthreads();

    float oxx[4], ozz[4], oxz[4];
#pragma unroll
    for (int j = 0; j < 4; ++j) {
        const int x  = x0 + j;
        const int xm = (x + NXC - 1) & (NXC - 1);
        const int xp = (x + 1) & (NXC - 1);

        const float vx0c = sm[x];
        const float vx0m = sm[xm];
        const float vx1c = sm[NXC + x];
        const float vz0c = sm[2 * NXC + x];
        const float vz0p = sm[2 * NXC + xp];
        const float vz1c = sm[3 * NXC + x];

        const float vx_x = vx0c - vx0m;
        const float vz_z = vz1c - vz0c;
        const float vx_z = vx1c - vx0c;
        const float vz_x = vz0p - vz0c;

        const float vpj = ((const float*)&vp4)[j];
        const float vsj = ((const float*)&vs4)[j];
        const float rj  = ((const float*)&rh4)[j];
        const float dj  = ((const float*)&d4)[j];

        const float mu   = rj * vsj * vsj;
        const float lam  = rj * vpj * vpj - 2.0f * mu;   // rho*(vp^2 - 2*vs^2)
        const float lp2m = lam + 2.0f * mu;

        const float c   = 0.5f * dt * dj;
        const float inv = 1.0f / (1.0f + c);
        const float omc = 1.0f - c;

        oxx[j] = inv * (dth * (lp2m * vx_x + lam * vz_z) + omc * ((const float*)&txx4)[j]);
        ozz[j] = inv * (dth * (lp2m * vz_z + lam * vx_x) + omc * ((const float*)&tzz4)[j]);
        oxz[j] = inv * (dth * mu * (vz_x + vx_z)         + omc * ((const float*)&txz4)[j]);
    }

    const size_t o = base_z + x0;
    *(float4*)(out + (size_t)2 * S + o) = make_float4(oxx[0], oxx[1], oxx[2], oxx[3]);
    *(float4*)(out + (size_t)3 * S + o) = make_float4(ozz[0], ozz[1], ozz[2], ozz[3]);
    *(float4*)(out + (size_t)4 * S + o) = make_float4(oxz[0], oxz[1], oxz[2], oxz[3]);
}

// ---------------------------------------------------------------------------
// Pass 2: velocity update from the new stresses (L2-resident from pass 1).
//   txx_x = y_txx[z,x+1]-y_txx[z,x]   txz_z = y_txz[z,x]-y_txz[z-1,x]
//   tzz_z = y_tzz[z,x]-y_tzz[z-1,x]   txz_x = y_txz[z,x]-y_txz[z,x-1]
// Writes y_vx -> out+0, y_vz -> out+S.
__global__ __launch_bounds__(256)
void velocity_kernel(const float* __restrict__ rho,
                     const float* __restrict__ vx, const float* __restrict__ vz,
                     const float* __restrict__ dtp, const float* __restrict__ hp,
                     const float* __restrict__ d,
                     const float* __restrict__ ytxx, const float* __restrict__ ytzz,
                     const float* __restrict__ ytxz,
                     float* __restrict__ out, int S) {
    __shared__ float sm[2 * NXC];   // [y_txx row z | y_txz row z]

    const int tid = threadIdx.x;
    const int z   = blockIdx.x & (NZC - 1);
    const int b   = blockIdx.x >> 10;
    const int x0  = tid << 2;
    const int zm  = (z + NZC - 1) & (NZC - 1);

    const size_t base_z  = ((size_t)b * NZC + z)  * NXC;
    const size_t base_zm = ((size_t)b * NZC + zm) * NXC;

    // Rows needing x-neighbors -> async LDS staging.
    async_load_b128((unsigned)((0 * NXC + x0) * 4), ytxx + base_z + x0, sm);
    async_load_b128((unsigned)((1 * NXC + x0) * 4), ytxz + base_z + x0, sm);

    // Center-only operands: direct B128 loads.
    const float4 txzm4 = *(const float4*)(ytxz + base_zm + x0);
    const float4 tzzc4 = *(const float4*)(ytzz + base_z  + x0);
    const float4 tzzm4 = *(const float4*)(ytzz + base_zm + x0);
    const float4 vx4   = *(const float4*)(vx   + base_z  + x0);
    const float4 vz4   = *(const float4*)(vz   + base_z  + x0);
    const size_t m = (size_t)z * NXC + x0;
    const float4 rh4 = *(const float4*)(rho + m);
    const float4 d4  = *(const float4*)(d   + m);
    const float  dt  = *dtp;
    const float  dth = dt / *hp;

    async_wait0();
    __syncthreads();

    float ovx[4], ovz[4];
#pragma unroll
    for (int j = 0; j < 4; ++j) {
        const int x  = x0 + j;
        const int xm = (x + NXC - 1) & (NXC - 1);
        const int xp = (x + 1) & (NXC - 1);

        const float txx_c = sm[x];
        const float txx_p = sm[xp];
        const float txz_c = sm[NXC + x];
        const float txz_m = sm[NXC + xm];

        const float txx_x  = txx_p - txx_c;
        const float txz_x  = txz_c - txz_m;
        const float txz_zd = txz_c - ((const float*)&txzm4)[j];
        const float tzz_zd = ((const float*)&tzzc4)[j] - ((const float*)&tzzm4)[j];

        const float rj = ((const float*)&rh4)[j];
        const float dj = ((const float*)&d4)[j];

        const float c    = 0.5f * dt * dj;
        const float inv  = 1.0f / (1.0f + c);
        const float omc  = 1.0f - c;
        const float dthr = dth / rj;

        ovx[j] = inv * (dthr * (txx_x + txz_zd) + omc * ((const float*)&vx4)[j]);
        ovz[j] = inv * (dthr * (txz_x + tzz_zd) + omc * ((const float*)&vz4)[j]);
    }

    const size_t o = base_z + x0;
    *(float4*)(out + o)             = make_float4(ovx[0], ovx[1], ovx[2], ovx[3]);
    *(float4*)(out + (size_t)S + o) = make_float4(ovz[0], ovz[1], ovz[2], ovz[3]);
}

// ---------------------------------------------------------------------------
extern "C" void kernel_launch(void* const* d_in, const int* in_sizes, int n_in,
                              void* d_out, int out_size, void* d_ws, size_t ws_size,
                              hipStream_t stream) {
    // setup_inputs order: vp, vs, rho, vx, vz, txx, tzz, txz, dt, h, d
    const float* vp  = (const float*)d_in[0];
    const float* vs  = (const float*)d_in[1];
    const float* rho = (const float*)d_in[2];
    const float* vx  = (const float*)d_in[3];
    const float* vz  = (const float*)d_in[4];
    const float* txx = (const float*)d_in[5];
    const float* tzz = (const float*)d_in[6];
    const float* txz = (const float*)d_in[7];
    const float* dtp = (const float*)d_in[8];
    const float* hp  = (const float*)d_in[9];
    const float* d   = (const float*)d_in[10];
    float* out = (float*)d_out;

    const int S = in_sizes[3];                 // B*NZ*NX
    const int B = S / (NZC * NXC);

    dim3 block(256);
    dim3 grid(B * NZC);                        // one (b, z) row per block

    stress_kernel<<<grid, block, 0, stream>>>(vp, vs, rho, vx, vz, txx, tzz, txz,
                                              dtp, hp, d, out, S);
    velocity_kernel<<<grid, block, 0, stream>>>(rho, vx, vz, dtp, hp, d,
                                                out + (size_t)2 * S,
                                                out + (size_t)3 * S,
                                                out + (size_t)4 * S,
                                                out, S);
}